// Model_9216999817972
// MI455X (gfx1250) — compile-verified
//
#include <hip/hip_runtime.h>
#include <hip/hip_bf16.h>

typedef float v2f __attribute__((ext_vector_type(2)));
typedef float v8f __attribute__((ext_vector_type(8)));

#define EPS 1e-5f

// ---------------------------------------------------------------------------
// Kernel 0: zero the BN statistics accumulators (224 floats, padded to 256)
// ---------------------------------------------------------------------------
__global__ void zero_stats_kernel(float* stats) {
    stats[threadIdx.x] = 0.0f;
}

// ---------------------------------------------------------------------------
// Kernel 1: conv1 (B,2,12,36) -> (B,16,12,36), expert-gathered, + BN stats
// one thread per output pixel (all 16 cout), LDS-reduced sum/sumsq
// ---------------------------------------------------------------------------
__global__ __launch_bounds__(256) void conv1_kernel(
    const float* __restrict__ x, const int* __restrict__ idx,
    const float* __restrict__ w1, const float* __restrict__ b1,
    float* __restrict__ out1, float* __restrict__ sumG, float* __restrict__ sqG,
    int B)
{
    __shared__ float ssum[16];
    __shared__ float ssq[16];
    if (threadIdx.x < 16) { ssum[threadIdx.x] = 0.0f; ssq[threadIdx.x] = 0.0f; }
    __syncthreads();

    const int HW = 12 * 36;
    int gid = blockIdx.x * blockDim.x + threadIdx.x;
    if (gid < B * HW) {
        int b  = gid / HW;
        int p  = gid - b * HW;
        int oh = p / 36, ow = p - (p / 36) * 36;
        int e  = idx[b];

        float patch[2][3][3];
        #pragma unroll
        for (int ci = 0; ci < 2; ++ci)
            #pragma unroll
            for (int ky = 0; ky < 3; ++ky)
                #pragma unroll
                for (int kx = 0; kx < 3; ++kx) {
                    int iy = oh + ky - 1, ix = ow + kx - 1;
                    patch[ci][ky][kx] = (iy >= 0 && iy < 12 && ix >= 0 && ix < 36)
                        ? x[((b * 2 + ci) * 12 + iy) * 36 + ix] : 0.0f;
                }

        const float* we = w1 + e * 16 * 18;   // (16,2,3,3)
        const float* be = b1 + e * 16;
        #pragma unroll
        for (int co = 0; co < 16; ++co) {
            float acc = be[co];
            const float* wc = we + co * 18;
            #pragma unroll
            for (int ci = 0; ci < 2; ++ci)
                #pragma unroll
                for (int ky = 0; ky < 3; ++ky)
                    #pragma unroll
                    for (int kx = 0; kx < 3; ++kx)
                        acc = fmaf(patch[ci][ky][kx], wc[(ci * 3 + ky) * 3 + kx], acc);
            out1[((b * 16 + co) * 12 + oh) * 36 + ow] = acc;
            atomicAdd(&ssum[co], acc);
            atomicAdd(&ssq[co], acc * acc);
        }
    }
    __syncthreads();
    if (threadIdx.x < 16) {
        atomicAdd(&sumG[threadIdx.x], ssum[threadIdx.x]);
        atomicAdd(&sqG[threadIdx.x],  ssq[threadIdx.x]);
    }
}

// ---------------------------------------------------------------------------
// Fused BN(batch-stats) + ReLU + maxpool(2,2,pad1).  One thread per pooled
// output element; stats finalized on the fly. (C,H,W,OH,OW are runtime args.)
// ---------------------------------------------------------------------------
__global__ __launch_bounds__(256) void bn_relu_pool_kernel(
    const float* __restrict__ in, float* __restrict__ outp,
    const float* __restrict__ sum, const float* __restrict__ sq,
    const float* __restrict__ gamma, const float* __restrict__ beta,
    int B, int C, int H, int W, int OH, int OW)
{
    int gid = blockIdx.x * blockDim.x + threadIdx.x;
    int total = B * C * OH * OW;
    if (gid >= total) return;
    int ow = gid % OW;
    int t  = gid / OW;
    int oh = t % OH;  t /= OH;
    int c  = t % C;
    int b  = t / C;

    float n     = (float)B * (float)H * (float)W;
    float mean  = sum[c] / n;
    float var   = sq[c] / n - mean * mean;
    float scale = gamma[c] * rsqrtf(var + EPS);
    float shift = beta[c] - mean * scale;

    const float* base = in + ((size_t)(b * C + c)) * H * W;
    float m = 0.0f;  // ReLU output >= 0 and every window has a valid element
    #pragma unroll
    for (int dy = 0; dy < 2; ++dy) {
        int ih = 2 * oh - 1 + dy;
        if (ih < 0 || ih >= H) continue;
        #pragma unroll
        for (int dx = 0; dx < 2; ++dx) {
            int iw = 2 * ow - 1 + dx;
            if (iw < 0 || iw >= W) continue;
            float v = fmaxf(fmaf(base[ih * W + iw], scale, shift), 0.0f);
            m = fmaxf(m, v);
        }
    }
    outp[gid] = m;
}

// ---------------------------------------------------------------------------
// Kernel 2: conv2 (B,16,7,19)->(B,32,7,19).  One block per sample: input tile
// (2128 f) + expert weights (4608 f) staged in LDS, 133 pixel threads.
// ---------------------------------------------------------------------------
__global__ __launch_bounds__(160) void conv2_kernel(
    const float* __restrict__ p1, const int* __restrict__ idx,
    const float* __restrict__ w2, const float* __restrict__ b2,
    float* __restrict__ out2, float* __restrict__ sumG, float* __restrict__ sqG)
{
    __shared__ float sin2[16 * 7 * 19];   // 2128
    __shared__ float wsh[32 * 144];       // 4608
    __shared__ float ssum[32];
    __shared__ float ssq[32];

    int b = blockIdx.x, tid = threadIdx.x;
    int e = idx[b];
    for (int i = tid; i < 2128; i += 160) sin2[i] = p1[b * 2128 + i];
    const float* we = w2 + e * 4608;
    for (int i = tid; i < 4608; i += 160) wsh[i] = we[i];
    if (tid < 32) { ssum[tid] = 0.0f; ssq[tid] = 0.0f; }
    __syncthreads();

    if (tid < 133) {
        int oh = tid / 19, ow = tid - (tid / 19) * 19;
        const float* be = b2 + e * 32;
        for (int co = 0; co < 32; ++co) {
            float acc = be[co];
            const float* wc = wsh + co * 144;
            for (int ci = 0; ci < 16; ++ci) {
                #pragma unroll
                for (int ky = 0; ky < 3; ++ky) {
                    int iy = oh + ky - 1;
                    if (iy < 0 || iy >= 7) continue;
                    #pragma unroll
                    for (int kx = 0; kx < 3; ++kx) {
                        int ix = ow + kx - 1;
                        if (ix < 0 || ix >= 19) continue;
                        acc = fmaf(sin2[(ci * 7 + iy) * 19 + ix],
                                   wc[(ci * 3 + ky) * 3 + kx], acc);
                    }
                }
            }
            out2[((b * 32 + co) * 7 + oh) * 19 + ow] = acc;
            atomicAdd(&ssum[co], acc);
            atomicAdd(&ssq[co], acc * acc);
        }
    }
    __syncthreads();
    if (tid < 32) {
        atomicAdd(&sumG[tid], ssum[tid]);
        atomicAdd(&sqG[tid],  ssq[tid]);
    }
}

// ---------------------------------------------------------------------------
// Kernel 3: conv3 (B,32,4,10)->(B,64,4,10).  One block (256 thr) per sample:
// tid -> (cout = tid>>2, row q = tid&3, 10 pixels).  Input tile in LDS,
// weight value broadcast over the 10-pixel inner loop.
// ---------------------------------------------------------------------------
__global__ __launch_bounds__(256) void conv3_kernel(
    const float* __restrict__ p2, const int* __restrict__ idx,
    const float* __restrict__ w3, const float* __restrict__ b3,
    float* __restrict__ out3, float* __restrict__ sumG, float* __restrict__ sqG)
{
    __shared__ float sin3[32 * 4 * 10];   // 1280
    __shared__ float ssum[64];
    __shared__ float ssq[64];

    int b = blockIdx.x, tid = threadIdx.x;
    int e = idx[b];
    for (int i = tid; i < 1280; i += 256) sin3[i] = p2[b * 1280 + i];
    if (tid < 64) { ssum[tid] = 0.0f; ssq[tid] = 0.0f; }
    __syncthreads();

    int co = tid >> 2;
    int q  = tid & 3;                      // output row oh == q
    float acc[10];
    float bias = b3[e * 64 + co];
    #pragma unroll
    for (int i = 0; i < 10; ++i) acc[i] = bias;

    const float* wc = w3 + ((size_t)e * 64 + co) * 288;  // (32,3,3)
    for (int ci = 0; ci < 32; ++ci) {
        #pragma unroll
        for (int ky = 0; ky < 3; ++ky) {
            int iy = q + ky - 1;
            bool yok = (iy >= 0 && iy < 4);
            #pragma unroll
            for (int kx = 0; kx < 3; ++kx) {
                float wv = wc[(ci * 3 + ky) * 3 + kx];
                if (!yok) continue;
                const float* row = sin3 + (ci * 4 + iy) * 10;
                #pragma unroll
                for (int i = 0; i < 10; ++i) {
                    int ix = i + kx - 1;
                    if (ix >= 0 && ix < 10) acc[i] = fmaf(row[ix], wv, acc[i]);
                }
            }
        }
    }
    float bs = 0.0f, bq = 0.0f;
    #pragma unroll
    for (int i = 0; i < 10; ++i) {
        out3[((b * 64 + co) * 4 + q) * 10 + i] = acc[i];
        bs += acc[i];
        bq += acc[i] * acc[i];
    }
    atomicAdd(&ssum[co], bs);
    atomicAdd(&ssq[co],  bq);
    __syncthreads();
    if (tid < 64) {
        atomicAdd(&sumG[tid], ssum[tid]);
        atomicAdd(&sqG[tid],  ssq[tid]);
    }
}

// ---------------------------------------------------------------------------
// Kernel 4: fused FC1 (WMMA f32 16x16x4) + ReLU + FC2.
// Each wave owns 16 batch rows x all 64 hidden cols (4 accumulator tiles),
// K = 1152 -> 288 v_wmma_f32_16x16x4_f32 steps per tile.
// FC2 folded in-register: h*fw2 summed with shfl_xor, out[b] written directly.
// ---------------------------------------------------------------------------
__global__ __launch_bounds__(256) void fc_fused_kernel(
    const float* __restrict__ X,     // (B,1152)
    const float* __restrict__ fw1,   // (64,1152)
    const float* __restrict__ fb1,   // (64)
    const float* __restrict__ fw2,   // (1,64)
    float* __restrict__ out, int B)
{
    const int K = 1152;
    int lane = threadIdx.x & 31;
    int wave = (blockIdx.x * blockDim.x + threadIdx.x) >> 5;
    int row0 = wave * 16;
    if (row0 >= B) return;                 // wave-uniform: EXEC stays all-ones

    int half = lane >> 4;                  // 0: K=k0,k0+1  1: K=k0+2,k0+3
    int l15  = lane & 15;

    const float* arow = X   + (size_t)(row0 + l15) * K + (half << 1);
    const float* brow = fw1 + (size_t)l15 * K + (half << 1);   // B = fw1^T tile

    v8f acc0 = {}, acc1 = {}, acc2 = {}, acc3 = {};
    for (int k0 = 0; k0 < K; k0 += 4) {
        if ((k0 & 63) == 0) __builtin_prefetch(arow + k0 + 128, 0, 1);
        v2f a  = *(const v2f*)(arow + k0);
        v2f b0 = *(const v2f*)(brow + k0);
        v2f b1 = *(const v2f*)(brow + 16 * K + k0);
        v2f b2 = *(const v2f*)(brow + 32 * K + k0);
        v2f b3 = *(const v2f*)(brow + 48 * K + k0);
        acc0 = __builtin_amdgcn_wmma_f32_16x16x4_f32(false, a, false, b0, (short)0, acc0, false, false);
        acc1 = __builtin_amdgcn_wmma_f32_16x16x4_f32(false, a, false, b1, (short)0, acc1, false, false);
        acc2 = __builtin_amdgcn_wmma_f32_16x16x4_f32(false, a, false, b2, (short)0, acc2, false, false);
        acc3 = __builtin_amdgcn_wmma_f32_16x16x4_f32(false, a, false, b3, (short)0, acc3, false, false);
    }

    // epilogue: h = relu(acc + fb1[col]); partial = h * fw2[col]; reduce cols
    float bias0 = fb1[l15],      bias1 = fb1[16 + l15];
    float bias2 = fb1[32 + l15], bias3 = fb1[48 + l15];
    float cf0 = fw2[l15],        cf1 = fw2[16 + l15];
    float cf2 = fw2[32 + l15],   cf3 = fw2[48 + l15];

    #pragma unroll
    for (int r = 0; r < 8; ++r) {
        float v = fmaxf(acc0[r] + bias0, 0.0f) * cf0
                + fmaxf(acc1[r] + bias1, 0.0f) * cf1
                + fmaxf(acc2[r] + bias2, 0.0f) * cf2
                + fmaxf(acc3[r] + bias3, 0.0f) * cf3;
        // reduce the 16 columns held in each lane-half (masks < 16 stay in-half)
        v += __shfl_xor(v, 1, 32);
        v += __shfl_xor(v, 2, 32);
        v += __shfl_xor(v, 4, 32);
        v += __shfl_xor(v, 8, 32);
        if (l15 == 0) out[row0 + r + (half << 3)] = v;  // D: VGPR r -> M=r / r+8
    }
}

// ---------------------------------------------------------------------------
extern "C" void kernel_launch(void* const* d_in, const int* in_sizes, int n_in,
                              void* d_out, int out_size, void* d_ws, size_t ws_size,
                              hipStream_t stream) {
    const float* x   = (const float*)d_in[0];
    const int*   idx = (const int*)  d_in[1];
    const float* w1  = (const float*)d_in[2];
    const float* b1  = (const float*)d_in[3];
    const float* w2  = (const float*)d_in[4];
    const float* b2  = (const float*)d_in[5];
    const float* w3  = (const float*)d_in[6];
    const float* b3  = (const float*)d_in[7];
    const float* g1  = (const float*)d_in[8];
    const float* be1 = (const float*)d_in[9];
    const float* g2  = (const float*)d_in[10];
    const float* be2 = (const float*)d_in[11];
    const float* g3  = (const float*)d_in[12];
    const float* be3 = (const float*)d_in[13];
    const float* fw1 = (const float*)d_in[14];
    const float* fb1 = (const float*)d_in[15];
    const float* fw2 = (const float*)d_in[16];
    float* out = (float*)d_out;

    const int B = 4096;

    // workspace layout (floats), with region reuse:
    //   stats[256] | A: conv outputs (28,311,552) | Bp: pool1 / flatten (8,716,288) | Cw: pool2 (5,242,880)
    float* stats = (float*)d_ws;
    float* s1 = stats,       *q1 = stats + 16;
    float* s2 = stats + 32,  *q2 = stats + 64;
    float* s3 = stats + 96,  *q3 = stats + 160;
    float* A  = stats + 256;
    float* Bp = A  + (size_t)B * 16 * 12 * 36;   // 28,311,552
    float* Cw = Bp + (size_t)B * 16 * 7 * 19;    //  8,716,288

    zero_stats_kernel<<<1, 256, 0, stream>>>(stats);

    // layer 1
    conv1_kernel<<<(B * 432 + 255) / 256, 256, 0, stream>>>(x, idx, w1, b1, A, s1, q1, B);
    bn_relu_pool_kernel<<<(B * 16 * 7 * 19 + 255) / 256, 256, 0, stream>>>(
        A, Bp, s1, q1, g1, be1, B, 16, 12, 36, 7, 19);

    // layer 2 (conv output reuses region A)
    conv2_kernel<<<B, 160, 0, stream>>>(Bp, idx, w2, b2, A, s2, q2);
    bn_relu_pool_kernel<<<(B * 32 * 4 * 10 + 255) / 256, 256, 0, stream>>>(
        A, Cw, s2, q2, g2, be2, B, 32, 7, 19, 4, 10);

    // layer 3 (conv output reuses region A; pooled/flattened reuses region Bp)
    conv3_kernel<<<B, 256, 0, stream>>>(Cw, idx, w3, b3, A, s3, q3);
    bn_relu_pool_kernel<<<(B * 64 * 3 * 6 + 255) / 256, 256, 0, stream>>>(
        A, Bp, s3, q3, g3, be3, B, 64, 4, 10, 3, 6);

    // fused FC1 (WMMA) + ReLU + FC2
    fc_fused_kernel<<<(B / 16) * 32 / 256, 256, 0, stream>>>(Bp, fw1, fb1, fw2, out, B);
}